// DocRedModel_30932354466339
// MI455X (gfx1250) — compile-verified
//
#include <hip/hip_runtime.h>
#include <math.h>

typedef __attribute__((ext_vector_type(2))) float v2f;
typedef __attribute__((ext_vector_type(8))) float v8f;

#define B_   4
#define M_   72
#define E_   24
#define R_   552
#define RP_  560              // 35 full 16-row tiles (zero-padded)
#define L_   1024
#define H_   1024
#define A_   16
#define EMB_ 768
#define BLK_ 64
#define OUT_ 768
#define NC_  97
#define N_   (B_ * R_)        // 2208 = 16 * 138
#define KBIL_ (EMB_ * BLK_)   // 49152

__device__ __forceinline__ v8f wmma4(v2f a, v2f b, v8f c) {
  // V_WMMA_F32_16X16X4_F32 : D = A(16x4) * B(4x16) + C(16x16), f32 accum
  return __builtin_amdgcn_wmma_f32_16x16x4_f32(
      /*neg_a=*/false, a, /*neg_b=*/false, b,
      /*c_mod=*/(short)0, c, /*reuse_a=*/false, /*reuse_b=*/false);
}

// ---------------------------------------------------------------------------
// 1) Per-segment logsumexp pool of ent_lhs -> ent_emb (B*E, H)
// ---------------------------------------------------------------------------
__global__ __launch_bounds__(256) void k_pool(const float* __restrict__ ent_lhs,
                                              const int* __restrict__ labels,
                                              float* __restrict__ ent_emb) {
  const int be = blockIdx.x;            // b*E + e
  const int b = be / E_, e = be % E_;
  __shared__ int lab[M_];
  for (int m = threadIdx.x; m < M_; m += blockDim.x) lab[m] = labels[b * M_ + m];
  __syncthreads();
  for (int d = threadIdx.x; d < H_; d += blockDim.x) {
    float mx = -INFINITY, s = 0.f;
    int cnt = 0;
    for (int m = 0; m < M_; ++m) {
      if (lab[m] == e) {
        float v = ent_lhs[((size_t)(b * M_ + m)) * H_ + d];
        if (v > mx) { s = s * __expf(mx - v) + 1.f; mx = v; }
        else        { s += __expf(v - mx); }
        ++cnt;
      }
    }
    ent_emb[(size_t)be * H_ + d] = (cnt > 0) ? (__logf(s) + mx) : 0.f;
  }
}

// ---------------------------------------------------------------------------
// 2) Per-segment mean of ent_to_seq_attn -> amean (B*E, A, L)
// ---------------------------------------------------------------------------
__global__ __launch_bounds__(256) void k_amean(const float* __restrict__ attn,
                                               const int* __restrict__ labels,
                                               float* __restrict__ amean) {
  const int bea = blockIdx.x;               // (b*E + e)*A + a
  const int a = bea % A_;
  const int be = bea / A_;
  const int e = be % E_, b = be / E_;
  __shared__ int lab[M_];
  for (int m = threadIdx.x; m < M_; m += blockDim.x) lab[m] = labels[b * M_ + m];
  __syncthreads();
  int cnt = 0;
  for (int m = 0; m < M_; ++m) cnt += (lab[m] == e);
  const float inv = (cnt > 0) ? 1.f / (float)cnt : 0.f;
  for (int l = threadIdx.x; l < L_; l += blockDim.x) {
    float s = 0.f;
    for (int m = 0; m < M_; ++m)
      if (lab[m] == e)
        s += attn[(((size_t)b * A_ + a) * M_ + m) * L_ + l];
    amean[(size_t)bea * L_ + l] = s * inv;
  }
}

// ---------------------------------------------------------------------------
// 3) ht_attn (zero-padded to RP_ rows per batch):
//    ht[b,r,l] = normalize_l( mean_a( amean[b,h,a,l] * amean[b,t,a,l] ) )
// ---------------------------------------------------------------------------
__global__ __launch_bounds__(256) void k_htattn(const float* __restrict__ amean,
                                                const int* __restrict__ hts,
                                                float* __restrict__ ht) {
  const int br = blockIdx.x;                // b*RP_ + r  (padded row space)
  const int b = br / RP_, r = br % RP_;
  float* dst = ht + (size_t)br * L_;
  if (r >= R_) {                            // zero pad rows -> unguarded GEMM A
    for (int l = threadIdx.x; l < L_; l += blockDim.x) dst[l] = 0.f;
    return;
  }
  const int n = b * R_ + r;
  const int h = hts[(size_t)n * 2 + 0];
  const int t = hts[(size_t)n * 2 + 1];
  const float* ha = amean + (size_t)(b * E_ + h) * A_ * L_;
  const float* ta = amean + (size_t)(b * E_ + t) * A_ * L_;
  float vals[L_ / 256];
  float local = 0.f;
#pragma unroll
  for (int i = 0; i < L_ / 256; ++i) {
    const int l = threadIdx.x + i * 256;
    float s = 0.f;
#pragma unroll
    for (int a = 0; a < A_; ++a) s += ha[a * L_ + l] * ta[a * L_ + l];
    s *= (1.f / (float)A_);
    vals[i] = s;
    local += s;
  }
  __shared__ float sred[256];
  sred[threadIdx.x] = local;
  __syncthreads();
  for (int s = 128; s > 0; s >>= 1) {
    if (threadIdx.x < s) sred[threadIdx.x] += sred[threadIdx.x + s];
    __syncthreads();
  }
  const float inv = 1.f / (sred[0] + 1e-5f);
#pragma unroll
  for (int i = 0; i < L_ / 256; ++i)
    dst[threadIdx.x + i * 256] = vals[i] * inv;
}

// ---------------------------------------------------------------------------
// 4) rel[b,r,:] = ht[b,r,:] @ seq_lhs[b,:,:]   (RP x L) @ (L x H)
//    One 16x16 D tile per wave; unconditional v2f A loads (padded A).
// ---------------------------------------------------------------------------
__global__ __launch_bounds__(128) void k_rel_gemm(const float* __restrict__ ht,
                                                  const float* __restrict__ seq,
                                                  float* __restrict__ rel) {
  const int b    = blockIdx.z;
  const int row0 = blockIdx.x * 16;
  const int wave = threadIdx.x >> 5;
  const int lane = threadIdx.x & 31;
  const int col0 = (blockIdx.y * 4 + wave) * 16;
  const int m  = lane & 15;
  const int kb = (lane >> 4) * 2;
  const float* Arow = ht + ((size_t)b * RP_ + row0 + m) * L_ + kb;
  const float* Bcol = seq + (size_t)b * L_ * H_ + (size_t)kb * H_ + col0 + m;
  v8f acc = {};
  for (int k0 = 0; k0 < L_; k0 += 4) {
    v2f a = *(const v2f*)(Arow + k0);            // global_load_b64
    v2f bv;
    bv.x = Bcol[(size_t)k0 * H_];
    bv.y = Bcol[(size_t)k0 * H_ + H_];
    acc = wmma4(a, bv, acc);
  }
#pragma unroll
  for (int v = 0; v < 8; ++v) {
    const int row = row0 + v + ((lane >> 4) << 3);
    if (row < R_)
      rel[((size_t)b * R_ + row) * H_ + col0 + (lane & 15)] = acc[v];
  }
}

// ---------------------------------------------------------------------------
// 5) out = tanh( [ent_emb[gather] | rel] @ W + bias ),  (N x 2048)@(2048 x 768)
//    Entity gather fused into the A-fragment load (no concat materialized).
// ---------------------------------------------------------------------------
__global__ __launch_bounds__(128) void k_proj_gemm(const float* __restrict__ ent_emb,
                                                   const float* __restrict__ rel,
                                                   const int* __restrict__ hts,
                                                   const float* __restrict__ W,
                                                   const float* __restrict__ bias,
                                                   float* __restrict__ outp,
                                                   int sel) {
  const int row0 = blockIdx.x * 16;         // N = 2208 divisible by 16
  const int wave = threadIdx.x >> 5;
  const int lane = threadIdx.x & 31;
  const int col0 = (blockIdx.y * 4 + wave) * 16;
  const int m  = lane & 15;
  const int kb = (lane >> 4) * 2;
  const int n = row0 + m;
  const int b = n / R_;
  const int e = hts[(size_t)n * 2 + sel];
  const float* Aent = ent_emb + (size_t)(b * E_ + e) * H_ + kb;
  const float* Arel = rel + (size_t)n * H_ + kb;
  const float* Bcol = W + (size_t)kb * EMB_ + col0 + m;
  v8f acc = {};
  for (int k0 = 0; k0 < H_; k0 += 4) {      // first half of K: gathered entity emb
    v2f a = *(const v2f*)(Aent + k0);
    v2f bv;
    bv.x = Bcol[(size_t)k0 * EMB_];
    bv.y = Bcol[(size_t)k0 * EMB_ + EMB_];
    acc = wmma4(a, bv, acc);
  }
  const float* Bcol2 = Bcol + (size_t)H_ * EMB_;
  for (int k0 = 0; k0 < H_; k0 += 4) {      // second half of K: rel
    v2f a = *(const v2f*)(Arel + k0);
    v2f bv;
    bv.x = Bcol2[(size_t)k0 * EMB_];
    bv.y = Bcol2[(size_t)k0 * EMB_ + EMB_];
    acc = wmma4(a, bv, acc);
  }
#pragma unroll
  for (int v = 0; v < 8; ++v) {
    const int row = row0 + v + ((lane >> 4) << 3);
    const int col = col0 + (lane & 15);
    outp[(size_t)row * EMB_ + col] = tanhf(acc[v] + bias[col]);
  }
}

// ---------------------------------------------------------------------------
// 6) embeds = bl @ W_bil + b_bil, bl[n, k*4096+i*64+j] = hs[n,k*64+i]*ts[n,k*64+j]
//    built on the fly from LDS-staged 16-row tiles (2 phases).
//    Since ka is even: hs index identical for ka,ka+1 (1 ds_load_b32) and
//    ts indices are j,j+1 contiguous (1 ds_load_b64 from padded row-major tile).
//    8 waves/WG share the LDS tile, each owns a 16-col tile of W_bil.
//    W_bil (151 MB) fits in the 192 MB L2 across re-streams.
// ---------------------------------------------------------------------------
#define HCOLS (EMB_ / 2)      // 384 staged hs columns per phase
#define TPAD  386             // ts row pitch: even (8B align) and 386 % 64 == 2
__global__ __launch_bounds__(256) void k_bilinear(const float* __restrict__ hs,
                                                  const float* __restrict__ ts,
                                                  const float* __restrict__ Wb,
                                                  const float* __restrict__ bb,
                                                  float* __restrict__ embeds) {
  __shared__ float shH[HCOLS][16];          // col-major: shH[c][r]   (24 KB)
  __shared__ float shT[16][TPAD];           // row-major padded       (~24.7 KB)
  const int row0 = blockIdx.x * 16;
  const int wave = threadIdx.x >> 5;
  const int lane = threadIdx.x & 31;
  const int col0 = (blockIdx.y * 8 + wave) * 16;
  const int m  = lane & 15;
  const int kb = (lane >> 4) * 2;
  const float* Bcol = Wb + col0 + m;
  v8f acc = {};
  for (int ph = 0; ph < 2; ++ph) {
    __syncthreads();                        // previous phase consumers done
    const int cbase = ph * HCOLS;
    for (int idx = threadIdx.x; idx < 16 * HCOLS; idx += 256) {
      const int c = idx % HCOLS;            // column fastest -> coalesced global
      const int r = idx / HCOLS;
      shH[c][r] = hs[(size_t)(row0 + r) * EMB_ + cbase + c];
      shT[r][c] = ts[(size_t)(row0 + r) * EMB_ + cbase + c];
    }
    __syncthreads();
    const int kbeg = ph * (KBIL_ / 2);
    const int kend = kbeg + (KBIL_ / 2);
    for (int k0 = kbeg; k0 < kend; k0 += 4) {
      const int ka = k0 + kb;               // even; ka and ka+1 share hs column
      const int hA = (ka >> 6) - cbase;                      // blk*64 + i
      const int tA = (((ka >> 12) << 6) | (ka & 63)) - cbase; // blk*64 + j (even)
      const float hv = shH[hA][m];                // ds_load_b32
      const v2f  tv = *(const v2f*)&shT[m][tA];   // ds_load_b64 (j, j+1)
      v2f a;
      a.x = hv * tv.x;
      a.y = hv * tv.y;
      v2f bv;
      bv.x = Bcol[(size_t)ka * OUT_];
      bv.y = Bcol[(size_t)ka * OUT_ + OUT_];
      acc = wmma4(a, bv, acc);
    }
  }
#pragma unroll
  for (int v = 0; v < 8; ++v) {
    const int row = row0 + v + ((lane >> 4) << 3);
    const int col = col0 + (lane & 15);
    embeds[(size_t)row * OUT_ + col] = acc[v] + bb[col];
  }
}

// ---------------------------------------------------------------------------
// 7) class_logits = embeds @ W_cls + b_cls ; binary = embeds @ W_bin + b_bin
// ---------------------------------------------------------------------------
__global__ __launch_bounds__(128) void k_cls(const float* __restrict__ embeds,
                                             const float* __restrict__ Wc,
                                             const float* __restrict__ bc,
                                             const float* __restrict__ Wbin,
                                             const float* __restrict__ bbin,
                                             float* __restrict__ cls_out,
                                             float* __restrict__ bin_out) {
  const int n = blockIdx.x;
  const int c = threadIdx.x;
  const float* en = embeds + (size_t)n * OUT_;
  if (c < NC_) {
    float s = bc[c];
    for (int k = 0; k < OUT_; ++k) s += en[k] * Wc[(size_t)k * NC_ + c];
    cls_out[(size_t)n * NC_ + c] = s;
  } else if (c == NC_) {
    float s = bbin[0];
    for (int k = 0; k < OUT_; ++k) s += en[k] * Wbin[k];
    bin_out[n] = s;
  }
}

extern "C" void kernel_launch(void* const* d_in, const int* in_sizes, int n_in,
                              void* d_out, int out_size, void* d_ws, size_t ws_size,
                              hipStream_t stream) {
  (void)in_sizes; (void)n_in; (void)out_size; (void)ws_size;
  const float* seq_lhs = (const float*)d_in[0];
  const float* ent_lhs = (const float*)d_in[1];
  const float* attn    = (const float*)d_in[2];
  const int*   labels  = (const int*)  d_in[3];
  const int*   hts     = (const int*)  d_in[4];
  const float* W_head  = (const float*)d_in[5];
  const float* b_head  = (const float*)d_in[6];
  const float* W_tail  = (const float*)d_in[7];
  const float* b_tail  = (const float*)d_in[8];
  const float* W_bil   = (const float*)d_in[9];
  const float* b_bil   = (const float*)d_in[10];
  const float* W_cls   = (const float*)d_in[11];
  const float* b_cls   = (const float*)d_in[12];
  const float* W_bin   = (const float*)d_in[13];
  const float* b_bin   = (const float*)d_in[14];

  float* ws      = (float*)d_ws;
  float* ent_emb = ws;                                    // B*E*H    =    98304
  float* amean   = ent_emb + (size_t)B_ * E_ * H_;        // B*E*A*L  =  1572864
  float* ht      = amean   + (size_t)B_ * E_ * A_ * L_;   // B*RP*L   =  2293760
  float* rel     = ht      + (size_t)B_ * RP_ * L_;       // N*H      =  2260992
  float* hsb     = rel     + (size_t)N_ * H_;             // N*EMB    =  1695744
  float* tsb     = hsb     + (size_t)N_ * EMB_;           // N*EMB    =  1695744

  float* embeds = (float*)d_out;
  float* clsl   = embeds + (size_t)N_ * OUT_;
  float* binl   = clsl   + (size_t)N_ * NC_;

  k_pool  <<<dim3(B_ * E_),       dim3(256), 0, stream>>>(ent_lhs, labels, ent_emb);
  k_amean <<<dim3(B_ * E_ * A_),  dim3(256), 0, stream>>>(attn, labels, amean);
  k_htattn<<<dim3(B_ * RP_),      dim3(256), 0, stream>>>(amean, hts, ht);
  // rel: 35 row tiles (560 padded rows), 64 col tiles, 4 waves/block, per-batch z
  k_rel_gemm<<<dim3(RP_ / 16, 16, B_), dim3(128), 0, stream>>>(ht, seq_lhs, rel);
  // projections: 138 row tiles x 48 col tiles, 4 waves/block
  k_proj_gemm<<<dim3(138, 12),    dim3(128), 0, stream>>>(ent_emb, rel, hts, W_head, b_head, hsb, 0);
  k_proj_gemm<<<dim3(138, 12),    dim3(128), 0, stream>>>(ent_emb, rel, hts, W_tail, b_tail, tsb, 1);
  // bilinear: 138 row tiles x 6 col groups (8 waves each -> 48 col tiles)
  k_bilinear<<<dim3(138, 6),      dim3(256), 0, stream>>>(hsb, tsb, W_bil, b_bil, embeds);
  k_cls   <<<dim3(N_),            dim3(128), 0, stream>>>(embeds, W_cls, b_cls, W_bin, b_bin, clsl, binl);
}